// RGCNClassifier_88648124990929
// MI455X (gfx1250) — compile-verified
//
#include <hip/hip_runtime.h>
#include <hip/hip_bf16.h>

#define NNODES  100000
#define NEDGES  1600000
#define NREL    8
#define HID     64
#define NGRAPHS 256
#define NTILES  (NNODES / 16)          // 6250 exact
#define WAVES_PER_BLOCK 8

typedef float v2f __attribute__((ext_vector_type(2)));
typedef float v8f __attribute__((ext_vector_type(8)));

// ---------------- workspace layout (float offsets) ----------------
#define OFF_INVC 0ull                      // R*N       =    800000
#define OFF_GCNT 800000ull                 // 256
#define OFF_GSUM 800256ull                 // 256*64    =     16384
#define OFF_X0   1000000ull                // N*64      =   6400000
#define OFF_X1   7400000ull                // N*64
#define OFF_AGG  13800000ull               // R*N*64    =  51200000
// total 65,000,000 floats = 260 MB

// ---------------- small kernels ----------------

__global__ void count_edges_kernel(const int* __restrict__ ei,
                                   const int* __restrict__ et,
                                   float* __restrict__ cnt) {
    int e = blockIdx.x * blockDim.x + threadIdx.x;
    if (e >= NEDGES) return;
    int seg = et[e] * NNODES + ei[NEDGES + e];
    unsafeAtomicAdd(&cnt[seg], 1.0f);
}

__global__ void invc_kernel(float* __restrict__ cnt) {
    int i = blockIdx.x * blockDim.x + threadIdx.x;
    if (i >= NREL * NNODES) return;
    cnt[i] = 1.0f / fmaxf(cnt[i], 1.0f);
}

__global__ void gcnt_kernel(const int* __restrict__ batch,
                            float* __restrict__ gcnt) {
    int n = blockIdx.x * blockDim.x + threadIdx.x;
    if (n >= NNODES) return;
    unsafeAtomicAdd(&gcnt[batch[n]], 1.0f);
}

// Per-wave-per-edge scatter: agg[etype*N+dst] += x[src] * invc (pre-scaled mean)
__global__ void scatter_kernel(const int* __restrict__ ei,
                               const int* __restrict__ et,
                               const float* __restrict__ invc,
                               const float* __restrict__ x,
                               float* __restrict__ agg) {
    unsigned wid = blockIdx.x * (blockDim.x >> 5) + (threadIdx.x >> 5);
    int lane = threadIdx.x & 31;
    if (wid >= NEDGES) return;
    int e    = (int)wid;
    int srcn = ei[e];
    int dstn = ei[NEDGES + e];
    int seg  = et[e] * NNODES + dstn;
    float w  = invc[seg];
    const float* xr = x + (size_t)srcn * HID;
    float* ar = agg + (size_t)seg * HID;
    int c = lane * 2;
    unsafeAtomicAdd(ar + c,     xr[c]     * w);
    unsafeAtomicAdd(ar + c + 1, xr[c + 1] * w);
}

// ---------------- WMMA GEMM kernels (fp32 16x16x4) ----------------
// A layout: lanes 0-15 hold rows M=0..15 (K=kc,kc+1 with half=0), lanes 16-31
// same rows with K=kc+2,kc+3.  B: VGPR0 = row K=kc (lanes0-15 N) / K=kc+2,
// VGPR1 = K=kc+1 / kc+3.  C/D: VGPR v -> row v + 8*half, lane%16 -> col.

// x0 = relu(concat(shape_emb[xn0], color_emb[xn1]) @ W_in + b_in)
__global__ __launch_bounds__(256)
void embed_gemm_kernel(const int* __restrict__ xn,
                       const float* __restrict__ shape_emb,
                       const float* __restrict__ color_emb,
                       const float* __restrict__ W_in,
                       const float* __restrict__ b_in,
                       float* __restrict__ out) {
    __shared__ float Bs[64 * 64];
    for (int i = threadIdx.x; i < 1024; i += blockDim.x)
        ((float4*)Bs)[i] = ((const float4*)W_in)[i];
    __syncthreads();

    int wave = threadIdx.x >> 5;
    int lane = threadIdx.x & 31;
    int tile = blockIdx.x * WAVES_PER_BLOCK + wave;
    if (tile >= NTILES) return;
    int nidx = lane & 15, half = lane >> 4;

    int gn = tile * 16 + nidx;                 // this lane's A row (node)
    const float* rowS = shape_emb + (size_t)xn[2 * gn]     * 32;
    const float* rowC = color_emb + (size_t)xn[2 * gn + 1] * 32;

    v8f acc[4] = {v8f{}, v8f{}, v8f{}, v8f{}};
#pragma unroll
    for (int k = 0; k < 16; ++k) {
        int kc = k * 4 + half * 2;             // even, never straddles 32
        v2f a;
        a.x = (kc     < 32) ? rowS[kc]     : rowC[kc - 32];
        a.y = (kc + 1 < 32) ? rowS[kc + 1] : rowC[kc + 1 - 32];
#pragma unroll
        for (int j = 0; j < 4; ++j) {
            v2f b;
            b.x = Bs[kc * 64 + j * 16 + nidx];
            b.y = Bs[(kc + 1) * 64 + j * 16 + nidx];
            acc[j] = __builtin_amdgcn_wmma_f32_16x16x4_f32(
                false, a, false, b, (short)0, acc[j], false, false);
        }
    }
#pragma unroll
    for (int j = 0; j < 4; ++j) {
        int col = j * 16 + nidx;
        float bv = b_in[col];
#pragma unroll
        for (int v = 0; v < 8; ++v) {
            int row = tile * 16 + v + half * 8;
            out[(size_t)row * HID + col] = fmaxf(acc[j][v] + bv, 0.0f);
        }
    }
}

// out = relu( sum_r agg[r] @ W_rel[r]  +  xin @ W_root  +  b )   (stacked 9 slabs)
__global__ __launch_bounds__(256)
void layer_gemm_kernel(const float* __restrict__ agg,
                       const float* __restrict__ xin,
                       const float* __restrict__ Wrel,
                       const float* __restrict__ Wroot,
                       const float* __restrict__ bias,
                       float* __restrict__ out) {
    __shared__ float Bs[64 * 64];
    int wave = threadIdx.x >> 5;
    int lane = threadIdx.x & 31;
    int tile = blockIdx.x * WAVES_PER_BLOCK + wave;
    int nidx = lane & 15, half = lane >> 4;

    v8f acc[4] = {v8f{}, v8f{}, v8f{}, v8f{}};
    for (int s = 0; s < 9; ++s) {
        const float* Bsrc = (s < 8) ? (Wrel + (size_t)s * 4096) : Wroot;
        for (int i = threadIdx.x; i < 1024; i += blockDim.x)
            ((float4*)Bs)[i] = ((const float4*)Bsrc)[i];
        __syncthreads();
        if (tile < NTILES) {
            const float* Abase = (s < 8) ? (agg + (size_t)s * NNODES * HID) : xin;
            const float* Arow  = Abase + (size_t)(tile * 16 + nidx) * HID;
#pragma unroll
            for (int k = 0; k < 16; ++k) {
                int kc = k * 4 + half * 2;
                v2f a; a.x = Arow[kc]; a.y = Arow[kc + 1];
#pragma unroll
                for (int j = 0; j < 4; ++j) {
                    v2f b;
                    b.x = Bs[kc * 64 + j * 16 + nidx];
                    b.y = Bs[(kc + 1) * 64 + j * 16 + nidx];
                    acc[j] = __builtin_amdgcn_wmma_f32_16x16x4_f32(
                        false, a, false, b, (short)0, acc[j], false, false);
                }
            }
        }
        __syncthreads();
    }
    if (tile < NTILES) {
#pragma unroll
        for (int j = 0; j < 4; ++j) {
            int col = j * 16 + nidx;
            float bv = bias[col];
#pragma unroll
            for (int v = 0; v < 8; ++v) {
                int row = tile * 16 + v + half * 8;
                out[(size_t)row * HID + col] = fmaxf(acc[j][v] + bv, 0.0f);
            }
        }
    }
}

// ---------------- pooling + classifier ----------------

__global__ void pool_kernel(const int* __restrict__ batch,
                            const float* __restrict__ x,
                            float* __restrict__ gsum) {
    unsigned t = blockIdx.x * blockDim.x + threadIdx.x;
    unsigned n = t >> 5;
    int lane = t & 31;
    if (n >= NNODES) return;
    int g = batch[n];
    int c = lane * 2;
    unsafeAtomicAdd(&gsum[(size_t)g * HID + c],     x[(size_t)n * HID + c]);
    unsafeAtomicAdd(&gsum[(size_t)g * HID + c + 1], x[(size_t)n * HID + c + 1]);
}

__global__ void final_kernel(const float* __restrict__ gsum,
                             const float* __restrict__ gcnt,
                             const float* __restrict__ W_out,
                             const float* __restrict__ b_out,
                             float* __restrict__ out) {
    int g = blockIdx.x * blockDim.x + threadIdx.x;
    if (g >= NGRAPHS) return;
    float inv = 1.0f / fmaxf(gcnt[g], 1.0f);
    float a0 = b_out[0], a1 = b_out[1];
#pragma unroll
    for (int k = 0; k < HID; ++k) {
        float p = gsum[(size_t)g * HID + k] * inv;
        a0 += p * W_out[k * 2 + 0];
        a1 += p * W_out[k * 2 + 1];
    }
    out[g * 2 + 0] = a0;
    out[g * 2 + 1] = a1;
}

// ---------------- launcher ----------------

extern "C" void kernel_launch(void* const* d_in, const int* in_sizes, int n_in,
                              void* d_out, int out_size, void* d_ws, size_t ws_size,
                              hipStream_t stream) {
    const int*   x_nodes   = (const int*)  d_in[0];
    const int*   edge_idx  = (const int*)  d_in[1];
    const int*   edge_type = (const int*)  d_in[2];
    const int*   batch     = (const int*)  d_in[3];
    const float* shape_emb = (const float*)d_in[4];
    const float* color_emb = (const float*)d_in[5];
    const float* W_in      = (const float*)d_in[6];
    const float* b_in      = (const float*)d_in[7];
    const float* W_rel1    = (const float*)d_in[8];
    const float* W_root1   = (const float*)d_in[9];
    const float* b1        = (const float*)d_in[10];
    const float* W_rel2    = (const float*)d_in[11];
    const float* W_root2   = (const float*)d_in[12];
    const float* b2        = (const float*)d_in[13];
    const float* W_out     = (const float*)d_in[14];
    const float* b_out     = (const float*)d_in[15];
    float* out = (float*)d_out;

    float* ws   = (float*)d_ws;
    float* invc = ws + OFF_INVC;
    float* gcnt = ws + OFF_GCNT;
    float* gsum = ws + OFF_GSUM;
    float* x0   = ws + OFF_X0;    // layer-0 features; reused as layer-2 output
    float* x1   = ws + OFF_X1;
    float* agg  = ws + OFF_AGG;

    // zero accumulators (graph-capturable async memsets)
    hipMemsetAsync(invc, 0, (size_t)NREL * NNODES * sizeof(float), stream);
    hipMemsetAsync(gcnt, 0, (size_t)(NGRAPHS + NGRAPHS * HID) * sizeof(float), stream); // gcnt+gsum contiguous
    hipMemsetAsync(agg,  0, (size_t)NREL * NNODES * HID * sizeof(float), stream);

    // per-(rel,dst) inverse counts (graph is identical for both conv layers)
    count_edges_kernel<<<(NEDGES + 255) / 256, 256, 0, stream>>>(edge_idx, edge_type, invc);
    invc_kernel<<<(NREL * NNODES + 255) / 256, 256, 0, stream>>>(invc);
    gcnt_kernel<<<(NNODES + 255) / 256, 256, 0, stream>>>(batch, gcnt);

    const int gemm_blocks = (NTILES + WAVES_PER_BLOCK - 1) / WAVES_PER_BLOCK; // 782
    const int scat_blocks = (NEDGES + WAVES_PER_BLOCK - 1) / WAVES_PER_BLOCK; // 200000

    // input featurization
    embed_gemm_kernel<<<gemm_blocks, 256, 0, stream>>>(x_nodes, shape_emb, color_emb,
                                                       W_in, b_in, x0);
    // conv layer 1: scatter-mean then stacked WMMA GEMM
    scatter_kernel<<<scat_blocks, 256, 0, stream>>>(edge_idx, edge_type, invc, x0, agg);
    layer_gemm_kernel<<<gemm_blocks, 256, 0, stream>>>(agg, x0, W_rel1, W_root1, b1, x1);

    // conv layer 2
    hipMemsetAsync(agg, 0, (size_t)NREL * NNODES * HID * sizeof(float), stream);
    scatter_kernel<<<scat_blocks, 256, 0, stream>>>(edge_idx, edge_type, invc, x1, agg);
    layer_gemm_kernel<<<gemm_blocks, 256, 0, stream>>>(agg, x1, W_rel2, W_root2, b2, x0);

    // graph mean-pool + classifier
    pool_kernel<<<((NNODES * 32) + 255) / 256, 256, 0, stream>>>(batch, x0, gsum);
    final_kernel<<<1, NGRAPHS, 0, stream>>>(gsum, gcnt, W_out, b_out, out);
}